// SeparatedTemporalGNN_30236569764193
// MI455X (gfx1250) — compile-verified
//
#include <hip/hip_runtime.h>
#include <hip/hip_bf16.h>

// ---------------------------------------------------------------------------
// SeparatedTemporalGNN on MI455X (gfx1250, wave32)
//
// Strategy: GEMM-then-scatter RGCN.
//   h = x @ Win + b
//   per layer:
//     out_s = h @ root_s + bias_s ; out_t = h @ root_t + bias_t   (WMMA GEMM)
//     for each relation r (7 spatial + 3 temporal):
//        Y = h @ W_r            (W_r = sum_b comp[r,b]*basis[b], WMMA GEMM)
//        out[dst] += Y[src] * (1/max(cnt[r,dst],1))   (fp32 atomic scatter)
//     h = residual + relu(LN([out_s|out_t] @ fuse_w + fuse_b))   (fused WMMA)
//   head: relu(h@W1+b1) @ W2 + b2
// All dense math uses V_WMMA_F32_16X16X4_F32 (exact fp32).
// ---------------------------------------------------------------------------

typedef __attribute__((ext_vector_type(2))) float v2f;
typedef __attribute__((ext_vector_type(8))) float v8f;

#define HID 128
#define RS_ 7
#define BS_ 4
#define RT_ 3
#define BT_ 3
#define NREL 10

__device__ inline void atomAddF(float* p, float v) {
  __hip_atomic_fetch_add(p, v, __ATOMIC_RELAXED, __HIP_MEMORY_SCOPE_AGENT);
}

// ---------------------------------------------------------------------------
// Generic WMMA fp32 GEMM:  C[N x (NT*16)] = A[N x KD] @ WT^T  (+bias, opt relu)
// WT is stored transposed: WT[n][k], n = output column. Block = 256 thr = 8
// waves; wave w computes rows [blk*128 + w*16, +16) x all NT*16 columns.
// ---------------------------------------------------------------------------
template <int KD, int NT, int ACT>
__global__ __launch_bounds__(256) void gemm_ws(const float* __restrict__ A,
                                               const float* __restrict__ WT,
                                               const float* __restrict__ bias,
                                               float* __restrict__ C, int ldc) {
  const int wave = threadIdx.x >> 5;
  const int lane = threadIdx.x & 31;
  const int ncol = lane & 15;
  const int half = lane >> 4;
  const int row0 = blockIdx.x * 128 + wave * 16;

  v8f acc[NT];
#pragma unroll
  for (int t = 0; t < NT; ++t)
#pragma unroll
    for (int i = 0; i < 8; ++i) acc[t][i] = 0.0f;

  // A fragment: lane(0-15)=row, VGPR0/1 = K0/K1; lanes 16-31 carry K2/K3.
  const float* arow = A + (size_t)(row0 + ncol) * KD + half * 2;
  const float* wrow[NT];
#pragma unroll
  for (int t = 0; t < NT; ++t)
    wrow[t] = WT + (size_t)(t * 16 + ncol) * KD + half * 2;

  for (int k = 0; k < KD; k += 4) {
    v2f a = *(const v2f*)(arow + k);
#pragma unroll
    for (int t = 0; t < NT; ++t) {
      v2f b = *(const v2f*)(wrow[t] + k);
      acc[t] = __builtin_amdgcn_wmma_f32_16x16x4_f32(false, a, false, b,
                                                     (short)0, acc[t], false,
                                                     false);
    }
  }

#pragma unroll
  for (int t = 0; t < NT; ++t) {
    float bv = bias ? bias[t * 16 + ncol] : 0.0f;
#pragma unroll
    for (int g = 0; g < 8; ++g) {
      float y = acc[t][g] + bv;
      if (ACT == 1) y = fmaxf(y, 0.0f);
      C[(size_t)(row0 + g + half * 8) * ldc + t * 16 + ncol] = y;
    }
  }
}

// ---------------------------------------------------------------------------
// Fused fuse-GEMM + LayerNorm + ReLU + residual:
//   hn = relu(LN([hs|ht] @ WT(fuse)^T + fb));  h = (first? 0 : h) + hn
// Each wave owns 16 complete rows -> LN reduces across a 16-lane half.
// ---------------------------------------------------------------------------
__global__ __launch_bounds__(256) void fuse_ln(const float* __restrict__ hs,
                                               const float* __restrict__ ht,
                                               const float* __restrict__ WT,
                                               const float* __restrict__ fb,
                                               const float* __restrict__ lg,
                                               const float* __restrict__ lb,
                                               float* __restrict__ h,
                                               int first) {
  const int wave = threadIdx.x >> 5;
  const int lane = threadIdx.x & 31;
  const int ncol = lane & 15;
  const int half = lane >> 4;
  const int row0 = blockIdx.x * 128 + wave * 16;
  const int mrow = row0 + ncol;
  const int koff = half * 2;

  v8f acc[8];
#pragma unroll
  for (int t = 0; t < 8; ++t)
#pragma unroll
    for (int i = 0; i < 8; ++i) acc[t][i] = 0.0f;

  const float* as = hs + (size_t)mrow * HID + koff;
  const float* at = ht + (size_t)mrow * HID + koff;

  for (int k = 0; k < HID; k += 4) {  // first 128 of K: spatial half
    v2f a = *(const v2f*)(as + k);
#pragma unroll
    for (int t = 0; t < 8; ++t) {
      v2f b = *(const v2f*)(WT + (size_t)(t * 16 + ncol) * 256 + koff + k);
      acc[t] = __builtin_amdgcn_wmma_f32_16x16x4_f32(false, a, false, b,
                                                     (short)0, acc[t], false,
                                                     false);
    }
  }
  for (int k = 0; k < HID; k += 4) {  // second 128 of K: temporal half
    v2f a = *(const v2f*)(at + k);
#pragma unroll
    for (int t = 0; t < 8; ++t) {
      v2f b =
          *(const v2f*)(WT + (size_t)(t * 16 + ncol) * 256 + 128 + koff + k);
      acc[t] = __builtin_amdgcn_wmma_f32_16x16x4_f32(false, a, false, b,
                                                     (short)0, acc[t], false,
                                                     false);
    }
  }

  float fbv[8], lgv[8], lbv[8];
#pragma unroll
  for (int t = 0; t < 8; ++t) {
    fbv[t] = fb[t * 16 + ncol];
    lgv[t] = lg[t * 16 + ncol];
    lbv[t] = lb[t * 16 + ncol];
  }
#pragma unroll
  for (int t = 0; t < 8; ++t)
#pragma unroll
    for (int g = 0; g < 8; ++g) acc[t][g] += fbv[t];

#pragma unroll
  for (int g = 0; g < 8; ++g) {
    // mean over 128 cols of row (row0 + g + half*8)
    float s = 0.0f;
#pragma unroll
    for (int t = 0; t < 8; ++t) s += acc[t][g];
    s += __shfl_xor(s, 8, 32);
    s += __shfl_xor(s, 4, 32);
    s += __shfl_xor(s, 2, 32);
    s += __shfl_xor(s, 1, 32);
    float mu = s * (1.0f / 128.0f);
    float q = 0.0f;
#pragma unroll
    for (int t = 0; t < 8; ++t) {
      float d = acc[t][g] - mu;
      q += d * d;
    }
    q += __shfl_xor(q, 8, 32);
    q += __shfl_xor(q, 4, 32);
    q += __shfl_xor(q, 2, 32);
    q += __shfl_xor(q, 1, 32);
    float rstd = rsqrtf(q * (1.0f / 128.0f) + 1e-5f);
    int grow = row0 + g + half * 8;
#pragma unroll
    for (int t = 0; t < 8; ++t) {
      float y = (acc[t][g] - mu) * rstd * lgv[t] + lbv[t];
      y = fmaxf(y, 0.0f);
      size_t idx = (size_t)grow * HID + t * 16 + ncol;
      h[idx] = (first ? 0.0f : h[idx]) + y;
    }
  }
}

// ---------------------------------------------------------------------------
// Edge kernels
// ---------------------------------------------------------------------------
__global__ void count_edges(const int* __restrict__ et,
                            const int* __restrict__ dst,
                            float* __restrict__ cnt, int N, int E) {
  int e = blockIdx.x * blockDim.x + threadIdx.x;
  if (e < E) atomAddF(&cnt[(size_t)et[e] * N + dst[e]], 1.0f);
}

__global__ void recip_cnt(float* __restrict__ cnt, int n) {
  int i = blockIdx.x * blockDim.x + threadIdx.x;
  if (i < n) cnt[i] = 1.0f / fmaxf(cnt[i], 1.0f);
}

// One wave per edge; lane l handles 4 of the 128 features.
__global__ __launch_bounds__(256) void scatter_rel(
    const float* __restrict__ Y, const int* __restrict__ src,
    const int* __restrict__ dst, const int* __restrict__ et,
    const float* __restrict__ icnt, int rel, float* __restrict__ out, int N,
    int E) {
  int wid = (blockIdx.x * blockDim.x + threadIdx.x) >> 5;
  int lane = threadIdx.x & 31;
  int nw = (gridDim.x * blockDim.x) >> 5;
  for (int e = wid; e < E; e += nw) {
    if (et[e] != rel) continue;
    int s = src[e], d = dst[e];
    float ic = icnt[(size_t)rel * N + d];
    const float4* yp = (const float4*)(Y + (size_t)s * HID);
    float4 v = yp[lane];
    float* op = out + (size_t)d * HID + lane * 4;
    atomAddF(op + 0, v.x * ic);
    atomAddF(op + 1, v.y * ic);
    atomAddF(op + 2, v.z * ic);
    atomAddF(op + 3, v.w * ic);
  }
}

// ---------------------------------------------------------------------------
// Weight prep
// ---------------------------------------------------------------------------
// WT[r][o][f] = sum_b comp[r,b] * basis[b][f][o]   (transposed relation weight)
__global__ void build_relW(const float* __restrict__ basis,
                           const float* __restrict__ comp,
                           float* __restrict__ WT, int R, int B) {
  int idx = blockIdx.x * blockDim.x + threadIdx.x;
  if (idx >= R * HID * HID) return;
  int f = idx & 127;
  int o = (idx >> 7) & 127;
  int r = idx >> 14;
  float s = 0.0f;
  for (int b = 0; b < B; ++b)
    s += comp[r * B + b] * basis[(size_t)b * HID * HID + f * HID + o];
  WT[(size_t)r * HID * HID + o * HID + f] = s;
}

// dst[Nc][K] = src[K][Nc]^T
__global__ void transpose_mat(const float* __restrict__ src,
                              float* __restrict__ dst, int K, int Nc) {
  int idx = blockIdx.x * blockDim.x + threadIdx.x;
  if (idx >= K * Nc) return;
  int n = idx % Nc;
  int k = idx / Nc;
  dst[(size_t)n * K + k] = src[idx];
}

// ---------------------------------------------------------------------------
extern "C" void kernel_launch(void* const* d_in, const int* in_sizes, int n_in,
                              void* d_out, int out_size, void* d_ws,
                              size_t ws_size, hipStream_t stream) {
  const float* x = (const float*)d_in[0];
  const int* eidx = (const int*)d_in[1];
  const int* etyp = (const int*)d_in[2];
  const float* w_in = (const float*)d_in[3];
  const float* b_in = (const float*)d_in[4];
  const float* sp_basis = (const float*)d_in[5];
  const float* sp_comp = (const float*)d_in[6];
  const float* sp_root = (const float*)d_in[7];
  const float* sp_bias = (const float*)d_in[8];
  const float* tp_basis = (const float*)d_in[9];
  const float* tp_comp = (const float*)d_in[10];
  const float* tp_root = (const float*)d_in[11];
  const float* tp_bias = (const float*)d_in[12];
  const float* fuse_w = (const float*)d_in[13];
  const float* fuse_b = (const float*)d_in[14];
  const float* ln_g = (const float*)d_in[15];
  const float* ln_b = (const float*)d_in[16];
  const float* head_w1 = (const float*)d_in[17];
  const float* head_b1 = (const float*)d_in[18];
  const float* head_w2 = (const float*)d_in[19];
  const float* head_b2 = (const float*)d_in[20];
  float* out = (float*)d_out;

  const int N = in_sizes[0] / 64;   // 131072
  const int E = in_sizes[1] / 2;    // 1500000
  const int* src = eidx;
  const int* dst = eidx + E;

  // workspace layout (floats)
  float* ws = (float*)d_ws;
  const size_t NH = (size_t)N * HID;
  float* h = ws;                 // [N,128]
  float* hs = ws + NH;           // [N,128]
  float* ht = ws + 2 * NH;       // [N,128]
  float* Y = ws + 3 * NH;        // [N,128] relation-projected feats / head h1
  float* cnt = ws + 4 * NH;      // [10,N] -> becomes 1/max(cnt,1)
  float* wt = cnt + (size_t)NREL * N;
  float* wt_in = wt;                       // [128][64]
  float* wt_rel = wt_in + 128 * 64;        // [10][128][128]
  float* wt_rs = wt_rel + NREL * HID * HID;// [128][128]
  float* wt_rt = wt_rs + HID * HID;        // [128][128]
  float* wt_fu = wt_rt + HID * HID;        // [128][256]
  float* wt_h1 = wt_fu + HID * 256;        // [128][128]
  float* wt_h2 = wt_h1 + HID * HID;        // [32][128]

  const int gemm_blocks = N / 128;

  // degree counts (static per launch) -> reciprocal
  hipMemsetAsync(cnt, 0, (size_t)NREL * N * sizeof(float), stream);
  count_edges<<<(E + 255) / 256, 256, 0, stream>>>(etyp, dst, cnt, N, E);
  recip_cnt<<<(NREL * N + 255) / 256, 256, 0, stream>>>(cnt, NREL * N);

  // static transposed weights
  transpose_mat<<<(64 * 128 + 255) / 256, 256, 0, stream>>>(w_in, wt_in, 64,
                                                            128);
  transpose_mat<<<(128 * 128 + 255) / 256, 256, 0, stream>>>(head_w1, wt_h1,
                                                             128, 128);
  transpose_mat<<<(128 * 32 + 255) / 256, 256, 0, stream>>>(head_w2, wt_h2,
                                                            128, 32);

  // input projection: h = x @ w_in + b_in
  gemm_ws<64, 8, 0><<<gemm_blocks, 256, 0, stream>>>(x, wt_in, b_in, h, HID);

  for (int l = 0; l < 3; ++l) {
    // per-layer relation weights (basis combo) + transposed roots / fuse
    build_relW<<<(RS_ * HID * HID + 255) / 256, 256, 0, stream>>>(
        sp_basis + (size_t)l * BS_ * HID * HID, sp_comp + (size_t)l * RS_ * BS_,
        wt_rel, RS_, BS_);
    build_relW<<<(RT_ * HID * HID + 255) / 256, 256, 0, stream>>>(
        tp_basis + (size_t)l * BT_ * HID * HID, tp_comp + (size_t)l * RT_ * BT_,
        wt_rel + (size_t)RS_ * HID * HID, RT_, BT_);
    transpose_mat<<<(HID * HID + 255) / 256, 256, 0, stream>>>(
        sp_root + (size_t)l * HID * HID, wt_rs, HID, HID);
    transpose_mat<<<(HID * HID + 255) / 256, 256, 0, stream>>>(
        tp_root + (size_t)l * HID * HID, wt_rt, HID, HID);
    transpose_mat<<<(256 * HID + 255) / 256, 256, 0, stream>>>(
        fuse_w + (size_t)l * 256 * HID, wt_fu, 256, HID);

    // root terms initialize the aggregation outputs
    gemm_ws<128, 8, 0><<<gemm_blocks, 256, 0, stream>>>(
        h, wt_rs, sp_bias + (size_t)l * HID, hs, HID);
    gemm_ws<128, 8, 0><<<gemm_blocks, 256, 0, stream>>>(
        h, wt_rt, tp_bias + (size_t)l * HID, ht, HID);

    // per-relation: project then mean-scatter
    for (int r = 0; r < NREL; ++r) {
      gemm_ws<128, 8, 0><<<gemm_blocks, 256, 0, stream>>>(
          h, wt_rel + (size_t)r * HID * HID, (const float*)nullptr, Y, HID);
      scatter_rel<<<2048, 256, 0, stream>>>(Y, src, dst, etyp, cnt, r,
                                            (r < RS_) ? hs : ht, N, E);
    }

    // fuse + layernorm + relu + residual
    fuse_ln<<<gemm_blocks, 256, 0, stream>>>(
        hs, ht, wt_fu, fuse_b + (size_t)l * HID, ln_g + (size_t)l * HID,
        ln_b + (size_t)l * HID, h, l == 0 ? 1 : 0);
  }

  // head
  gemm_ws<128, 8, 1><<<gemm_blocks, 256, 0, stream>>>(h, wt_h1, head_b1, Y,
                                                      HID);
  gemm_ws<128, 2, 0><<<gemm_blocks, 256, 0, stream>>>(Y, wt_h2, head_b2, out,
                                                      32);
}